// CrossNonLocalBlock_32229434589814
// MI455X (gfx1250) — compile-verified
//
#include <hip/hip_runtime.h>

// ---------------------------------------------------------------------------
// CrossNonLocalBlock on MI455X (gfx1250), f32 end-to-end via V_WMMA_F32_16X16X4_F32.
// B=128, C=512, IC=256, N=H*W=256. All heavy math as WMMA GEMMs.
// ---------------------------------------------------------------------------

#define BB   128
#define CC   512
#define ICC  256
#define NN_  256

typedef float f32x2 __attribute__((ext_vector_type(2)));
typedef float f32x8 __attribute__((ext_vector_type(8)));

// ---------------------------------------------------------------------------
// Generic tiled WMMA GEMM:  D[b] (M x Nn) = A (M x K) @ Bmat[b] (K x Nn)
//   TRANS_A : A stored K-major as [K][M] (used for att = T^T @ P)
//   A_BATCH : A batched with stride M*K
//   BIAS    : += bias[m]
//   RELU    : max(.,0) epilogue
//   ACC     : += existing D (accumulate across launches)
//   RES     : += res[b] (final residual)
// Block: 256 threads (8 wave32). Block tile 64(M) x 128(N), K-step 32.
// Each wave computes a 32x32 slab = 4 WMMA accumulators; both LDS tiles are
// stored K-minor with an EVEN pad (pitch 34) so every fragment is one
// 8B-aligned ds_load_b64 landing in an even VGPR pair (no packing movs).
// ---------------------------------------------------------------------------
template<bool TRANS_A, bool A_BATCH, bool BIAS, bool RELU, bool ACC, bool RES>
__global__ __launch_bounds__(256)
void gemm_wmma_k(const float* __restrict__ A, const float* __restrict__ Bmat,
                 const float* __restrict__ bias, const float* __restrict__ res,
                 float* __restrict__ D, int M, int Nn, int K)
{
    constexpr int BM = 64, BN = 128, BK = 32;
    constexpr int PK = BK + 2;          // even pad: 8B-aligned rows, no bank clash
    __shared__ float As[BM][PK];        // As[m][k]
    __shared__ float Bs[BN][PK];        // Bs[n][k]  (B staged transposed)

    const int b    = blockIdx.z;
    const int m0   = blockIdx.y * BM;
    const int n0   = blockIdx.x * BN;
    const int t    = threadIdx.x;
    const int lane = t & 31;            // wave32
    const int wv   = t >> 5;            // 0..7
    const int wm   = (wv & 1) * 32;     // wave M offset in block tile
    const int wn   = (wv >> 1) * 32;    // wave N offset in block tile

    const float* Ab = A + (A_BATCH ? (size_t)b * (size_t)M * (size_t)K : (size_t)0);
    const float* Bb = Bmat + (size_t)b * (size_t)K * (size_t)Nn;

    f32x8 c00 = {}, c01 = {}, c10 = {}, c11 = {};

    const int la = lane & 15;           // A: M row / B: N col within 16-block
    const int kk = (lane >> 4) * 2;     // K pair within 4-slice

    for (int ko = 0; ko < K; ko += BK) {
        // ---- stage A tile (float4 global loads, coalesced) ----
        if (TRANS_A) {                       // A memory is [K][M], m contiguous
            #pragma unroll
            for (int i = 0; i < 2; ++i) {
                int lin = t + i * 256;       // 512 float4 slots
                int m4  = (lin & 15) * 4;
                int kq  = lin >> 4;          // 0..31
                float4 v = *reinterpret_cast<const float4*>(
                    Ab + (size_t)(ko + kq) * M + (m0 + m4));
                As[m4 + 0][kq] = v.x; As[m4 + 1][kq] = v.y;
                As[m4 + 2][kq] = v.z; As[m4 + 3][kq] = v.w;
            }
        } else {                             // A memory is [M][K], k contiguous
            #pragma unroll
            for (int i = 0; i < 2; ++i) {
                int lin = t + i * 256;
                int k4  = (lin & 7) * 4;
                int mm  = lin >> 3;          // 0..63
                float4 v = *reinterpret_cast<const float4*>(
                    Ab + (size_t)(m0 + mm) * K + (ko + k4));
                As[mm][k4 + 0] = v.x; As[mm][k4 + 1] = v.y;
                As[mm][k4 + 2] = v.z; As[mm][k4 + 3] = v.w;
            }
        }
        // ---- stage B tile transposed: Bs[n][k] <- B[k][n] ----
        #pragma unroll
        for (int i = 0; i < 4; ++i) {
            int lin = t + i * 256;           // 1024 float4 slots
            int n4  = (lin & 31) * 4;        // 0..124
            int kq  = lin >> 5;              // 0..31
            float4 v = *reinterpret_cast<const float4*>(
                Bb + (size_t)(ko + kq) * Nn + (n0 + n4));
            Bs[n4 + 0][kq] = v.x; Bs[n4 + 1][kq] = v.y;
            Bs[n4 + 2][kq] = v.z; Bs[n4 + 3][kq] = v.w;
        }
        // ---- prefetch next K-tile toward the WGP (global_prefetch_b8) ----
        if (ko + BK < K) {
            {
                int n4 = (t & 31) * 4, kq = t >> 5;
                __builtin_prefetch(Bb + (size_t)(ko + BK + kq) * Nn + (n0 + n4), 0, 0);
            }
            if (TRANS_A) {
                int m4 = (t & 15) * 4, kq = t >> 4;
                __builtin_prefetch(Ab + (size_t)(ko + BK + kq) * M + (m0 + m4), 0, 0);
            } else {
                int k4 = (t & 7) * 4, mm = t >> 3;
                __builtin_prefetch(Ab + (size_t)(m0 + mm) * K + (ko + BK + k4), 0, 0);
            }
        }
        __syncthreads();

        // ---- 8 x (16x16x4) K-slices; 4 WMMA per slice, 4 ds_load_b64 each ----
        #pragma unroll
        for (int k = 0; k < BK; k += 4) {
            f32x2 a0 = *reinterpret_cast<const f32x2*>(&As[wm + la     ][k + kk]);
            f32x2 a1 = *reinterpret_cast<const f32x2*>(&As[wm + 16 + la][k + kk]);
            f32x2 b0 = *reinterpret_cast<const f32x2*>(&Bs[wn + la     ][k + kk]);
            f32x2 b1 = *reinterpret_cast<const f32x2*>(&Bs[wn + 16 + la][k + kk]);
            c00 = __builtin_amdgcn_wmma_f32_16x16x4_f32(false, a0, false, b0,
                                                        (short)0, c00, false, false);
            c01 = __builtin_amdgcn_wmma_f32_16x16x4_f32(false, a0, false, b1,
                                                        (short)0, c01, false, false);
            c10 = __builtin_amdgcn_wmma_f32_16x16x4_f32(false, a1, false, b0,
                                                        (short)0, c10, false, false);
            c11 = __builtin_amdgcn_wmma_f32_16x16x4_f32(false, a1, false, b1,
                                                        (short)0, c11, false, false);
        }
        __syncthreads();
    }

    // ---- epilogue: C/D frag layout = VGPR r -> M=r (+8 for lanes 16..31) ----
    const int rsel = (lane >> 4) << 3;
    float* Db = D + (size_t)b * (size_t)M * (size_t)Nn;
    const float* Rb = RES ? (res + (size_t)b * (size_t)M * (size_t)Nn) : nullptr;

    #pragma unroll
    for (int mi = 0; mi < 2; ++mi) {
        const f32x8 cr0 = mi ? c10 : c00;
        const f32x8 cr1 = mi ? c11 : c01;
        const int mgb = m0 + wm + 16 * mi + rsel;
        #pragma unroll
        for (int r = 0; r < 8; ++r) {
            int mg = mgb + r;
            float v0 = cr0[r], v1 = cr1[r];
            if (BIAS) { float bv = bias[mg]; v0 += bv; v1 += bv; }
            if (RELU) { v0 = fmaxf(v0, 0.f); v1 = fmaxf(v1, 0.f); }
            size_t i0 = (size_t)mg * Nn + (n0 + wn + la);
            size_t i1 = (size_t)mg * Nn + (n0 + wn + 16 + la);
            if (ACC) { v0 += Db[i0]; v1 += Db[i1]; }
            if (RES) { v0 += Rb[i0]; v1 += Rb[i1]; }
            Db[i0] = v0; Db[i1] = v1;
        }
    }
}

// ---------------------------------------------------------------------------
// Degree normalization: att holds acc = T^T P + P^T T (att_sym = 0.5*acc).
// d[n] = 0.5 * sum_m acc[n][m];  dn = d!=0 ? rsqrt(d) : 0.  One wave per row.
// ---------------------------------------------------------------------------
__global__ __launch_bounds__(256)
void degree_k(const float* __restrict__ att, float* __restrict__ dn)
{
    int row  = blockIdx.x * 8 + (threadIdx.x >> 5);   // row in [0, B*N)
    int lane = threadIdx.x & 31;
    const float* p = att + (size_t)row * NN_;
    float s = 0.f;
    #pragma unroll
    for (int i = 0; i < NN_ / 32; ++i) s += p[lane + i * 32];
    #pragma unroll
    for (int off = 16; off > 0; off >>= 1) s += __shfl_xor(s, off, 32);
    if (lane == 0) {
        float d = 0.5f * s;
        dn[row] = (d != 0.f) ? rsqrtf(d) : 0.f;
    }
}

// f[b][n][m] = 0.5 * acc[b][n][m] * dn[b*N+n] * dn[b*N+m]   (in place)
__global__ __launch_bounds__(256)
void scale_k(float* __restrict__ att, const float* __restrict__ dn)
{
    size_t idx = (size_t)blockIdx.x * 256 + threadIdx.x;   // over B*N*N
    size_t m   = idx & (NN_ - 1);
    size_t row = idx >> 8;            // b*N + n
    size_t b   = row >> 8;
    att[idx] = 0.5f * att[idx] * dn[row] * dn[b * NN_ + m];
}

// Per-channel batch stats over (B, N): mean + rsqrt(var+eps). 1 block / channel.
__global__ __launch_bounds__(256)
void bnstats_k(const float* __restrict__ S, float* __restrict__ mean,
               float* __restrict__ rstd)
{
    int c = blockIdx.x;
    __shared__ float ssum[256], ssq[256];
    float s = 0.f, q = 0.f;
    for (int i = threadIdx.x; i < BB * NN_; i += 256) {
        int b = i >> 8, n = i & (NN_ - 1);
        float v = S[((size_t)b * CC + c) * NN_ + n];
        s += v; q += v * v;
    }
    ssum[threadIdx.x] = s; ssq[threadIdx.x] = q;
    __syncthreads();
    for (int off = 128; off > 0; off >>= 1) {
        if (threadIdx.x < off) {
            ssum[threadIdx.x] += ssum[threadIdx.x + off];
            ssq[threadIdx.x]  += ssq[threadIdx.x + off];
        }
        __syncthreads();
    }
    if (threadIdx.x == 0) {
        const float inv = 1.0f / (float)(BB * NN_);
        float mu  = ssum[0] * inv;
        float var = ssq[0] * inv - mu * mu;   // biased variance (training BN)
        mean[c] = mu;
        rstd[c] = rsqrtf(var + 1e-5f);
    }
}

// z = BN1(S1) + BN2(S2) + XS   (elementwise over [B,C,N])
__global__ __launch_bounds__(256)
void zfuse_k(const float* __restrict__ S1, const float* __restrict__ S2,
             const float* __restrict__ XS,
             const float* __restrict__ mu1, const float* __restrict__ rs1,
             const float* __restrict__ g1,  const float* __restrict__ b1,
             const float* __restrict__ mu2, const float* __restrict__ rs2,
             const float* __restrict__ g2,  const float* __restrict__ b2,
             float* __restrict__ z)
{
    size_t idx = (size_t)blockIdx.x * 256 + threadIdx.x;   // over B*C*N
    size_t c   = (idx >> 8) & (CC - 1);
    float v1 = (S1[idx] - mu1[c]) * rs1[c] * g1[c] + b1[c];
    float v2 = (S2[idx] - mu2[c]) * rs2[c] * g2[c] + b2[c];
    z[idx] = v1 + v2 + XS[idx];
}

// ---------------------------------------------------------------------------
// Host-side orchestration
// ---------------------------------------------------------------------------
extern "C" void kernel_launch(void* const* d_in, const int* in_sizes, int n_in,
                              void* d_out, int out_size, void* d_ws, size_t ws_size,
                              hipStream_t stream)
{
    (void)in_sizes; (void)n_in; (void)out_size; (void)ws_size;

    const float* x     = (const float*)d_in[0];
    const float* ob    = (const float*)d_in[1];
    const float* od    = (const float*)d_in[2];
    const float* gx_w  = (const float*)d_in[3];
    const float* gx_b  = (const float*)d_in[4];
    const float* gb_w  = (const float*)d_in[5];
    const float* gb_b  = (const float*)d_in[6];
    const float* gd_w  = (const float*)d_in[7];
    const float* gd_b  = (const float*)d_in[8];
    const float* t_w   = (const float*)d_in[9];
    const float* p_w   = (const float*)d_in[10];
    const float* Wx_w  = (const float*)d_in[11];
    const float* Wx_b  = (const float*)d_in[12];
    const float* Wb_w  = (const float*)d_in[13];
    const float* Wb_b  = (const float*)d_in[14];
    const float* Wd_w  = (const float*)d_in[15];
    const float* Wd_b  = (const float*)d_in[16];
    const float* Wxb_w = (const float*)d_in[17];
    const float* Wxb_b = (const float*)d_in[18];
    const float* Wxd_w = (const float*)d_in[19];
    const float* Wxd_b = (const float*)d_in[20];
    const float* bn1_g = (const float*)d_in[21];
    const float* bn1_b = (const float*)d_in[22];
    const float* bn2_g = (const float*)d_in[23];
    const float* bn2_b = (const float*)d_in[24];
    const float* out_w = (const float*)d_in[25];
    const float* out_b = (const float*)d_in[26];
    float* out = (float*)d_out;

    // ---- workspace layout (floats); ~436 MB total ----
    const size_t BICN = (size_t)BB * ICC * NN_;   // 8,388,608
    const size_t BNN  = (size_t)BB * NN_ * NN_;   // 8,388,608
    const size_t BCN  = (size_t)BB * CC  * NN_;   // 16,777,216
    float* ws  = (float*)d_ws;
    float* T   = ws;                  // [B,IC,N]
    float* P   = T  + BICN;           // [B,IC,N]
    float* fx  = P  + BICN;           // [B,N,N]
    float* fb  = fx + BNN;
    float* fd  = fb + BNN;
    float* dnb = fd + BNN;            // [B,N]
    float* G   = dnb + (size_t)BB * NN_;  // [B,IC,N]
    float* Hb  = G  + BICN;           // [B,IC,N]
    float* S1  = Hb + BICN;           // [B,C,N]
    float* S2  = S1 + BCN;
    float* XS  = S2 + BCN;
    float* st  = XS + BCN;            // 4*512 stats
    float* mu1 = st, *rs1 = st + CC, *mu2 = st + 2 * CC, *rs2 = st + 3 * CC;
    float* z   = T;                   // reuse T+P region (exactly BCN floats)

    const dim3 blk(256);
    const dim3 gIC(NN_ / 128, ICC / 64, BB);  // M=256 GEMMs: (2,4,128)
    const dim3 gC (NN_ / 128, CC  / 64, BB);  // M=512 GEMMs: (2,8,128)
    const dim3 gNN(NN_ / 128, NN_ / 64, BB);  // att GEMMs:   (2,4,128)

    // ---- affinity matrices f_x, f_b, f_d (T/P buffers reused per branch) ----
    const float* srcs[3] = { x, ob, od };
    float*       fs[3]   = { fx, fb, fd };
    for (int i = 0; i < 3; ++i) {
        const float* in = srcs[i];
        float* f = fs[i];
        // T = relu(t_w @ in), P = relu(p_w @ in)          [IC,N], K=C
        gemm_wmma_k<false,false,false,true,false,false><<<gIC, blk, 0, stream>>>(
            t_w, in, nullptr, nullptr, T, ICC, NN_, CC);
        gemm_wmma_k<false,false,false,true,false,false><<<gIC, blk, 0, stream>>>(
            p_w, in, nullptr, nullptr, P, ICC, NN_, CC);
        // acc = T^T @ P ; acc += P^T @ T   (= att + att^T)   [N,N], K=IC
        gemm_wmma_k<true,true,false,false,false,false><<<gNN, blk, 0, stream>>>(
            T, P, nullptr, nullptr, f, NN_, NN_, ICC);
        gemm_wmma_k<true,true,false,false,true,false><<<gNN, blk, 0, stream>>>(
            P, T, nullptr, nullptr, f, NN_, NN_, ICC);
        // degree normalize (0.5 factor folded in)
        degree_k<<<(BB * NN_) / 8, blk, 0, stream>>>(f, dnb);
        scale_k<<<BNN / 256, blk, 0, stream>>>(f, dnb);
    }

    // ---- S1 = Wd@(G_d f_d) + Wxb@(G_x f_b) ; S2 = Wb@(G_b f_b) + Wxd@(G_x f_d)
    //      XS = Wx@(G_x f_x)   (one G and one H buffer, sequenced) ----
    // G_d
    gemm_wmma_k<false,false,true,false,false,false><<<gIC, blk, 0, stream>>>(
        gd_w, od, gd_b, nullptr, G, ICC, NN_, CC);
    gemm_wmma_k<false,true,false,false,false,false><<<gIC, blk, 0, stream>>>(
        G, fd, nullptr, nullptr, Hb, ICC, NN_, NN_);
    gemm_wmma_k<false,false,true,false,false,false><<<gC, blk, 0, stream>>>(
        Wd_w, Hb, Wd_b, nullptr, S1, CC, NN_, ICC);
    // G_b
    gemm_wmma_k<false,false,true,false,false,false><<<gIC, blk, 0, stream>>>(
        gb_w, ob, gb_b, nullptr, G, ICC, NN_, CC);
    gemm_wmma_k<false,true,false,false,false,false><<<gIC, blk, 0, stream>>>(
        G, fb, nullptr, nullptr, Hb, ICC, NN_, NN_);
    gemm_wmma_k<false,false,true,false,false,false><<<gC, blk, 0, stream>>>(
        Wb_w, Hb, Wb_b, nullptr, S2, CC, NN_, ICC);
    // G_x (kept for three aggregations)
    gemm_wmma_k<false,false,true,false,false,false><<<gIC, blk, 0, stream>>>(
        gx_w, x, gx_b, nullptr, G, ICC, NN_, CC);
    gemm_wmma_k<false,true,false,false,false,false><<<gIC, blk, 0, stream>>>(
        G, fb, nullptr, nullptr, Hb, ICC, NN_, NN_);
    gemm_wmma_k<false,false,true,false,true,false><<<gC, blk, 0, stream>>>(
        Wxb_w, Hb, Wxb_b, nullptr, S1, CC, NN_, ICC);          // S1 += cross
    gemm_wmma_k<false,true,false,false,false,false><<<gIC, blk, 0, stream>>>(
        G, fd, nullptr, nullptr, Hb, ICC, NN_, NN_);
    gemm_wmma_k<false,false,true,false,true,false><<<gC, blk, 0, stream>>>(
        Wxd_w, Hb, Wxd_b, nullptr, S2, CC, NN_, ICC);          // S2 += cross
    gemm_wmma_k<false,true,false,false,false,false><<<gIC, blk, 0, stream>>>(
        G, fx, nullptr, nullptr, Hb, ICC, NN_, NN_);
    gemm_wmma_k<false,false,true,false,false,false><<<gC, blk, 0, stream>>>(
        Wx_w, Hb, Wx_b, nullptr, XS, CC, NN_, ICC);            // x_self

    // ---- BN stats + fused normalize/sum ----
    bnstats_k<<<CC, blk, 0, stream>>>(S1, mu1, rs1);
    bnstats_k<<<CC, blk, 0, stream>>>(S2, mu2, rs2);
    zfuse_k<<<BCN / 256, blk, 0, stream>>>(S1, S2, XS,
                                           mu1, rs1, bn1_g, bn1_b,
                                           mu2, rs2, bn2_g, bn2_b, z);

    // ---- out = out_w @ z + out_b + x ----
    gemm_wmma_k<false,false,true,false,false,true><<<gC, blk, 0, stream>>>(
        out_w, z, out_b, x, out, CC, NN_, CC);
}